// PixelwiseContrastiveLoss_28587302322528
// MI455X (gfx1250) — compile-verified
//
#include <hip/hip_runtime.h>
#include <hip/hip_bf16.h>
#include <math.h>

// ---- problem constants (from reference) ----
#define BATCH 2
#define DD 128          // feature dim
#define CC 19           // classes
#define PP 3600         // anchor pixels (60*60)  = 48 * 75
#define NN 57600        // negatives    (240*240) = 384 * 150
#define NBLK 384        // N per block = 8 waves * 48
#define PBLK 48         // P per block = 3 tiles of 16
#define INV_TEMP 10.0f  // 1 / 0.1
#define GAMMA_C 0.75f
#define EPS_C 1e-8f

typedef float v2f __attribute__((ext_vector_type(2)));
typedef float v8f __attribute__((ext_vector_type(8)));

// xor-butterfly within the 16-lane half via ds_swizzle (and=0x1F, or=0, xor=m)
#define SWZ_ADD(v, imm) \
    v += __int_as_float(__builtin_amdgcn_ds_swizzle(__float_as_int(v), imm))

// ------------------------------------------------------------------
// Kernel 1: per-anchor prep: labels1 (argmax_c pl1), pos_exp, mask1,
//           zero neg_logits and the 4 reduction accumulators.
// ------------------------------------------------------------------
__global__ void prep_kernel(const float* __restrict__ pl1,
                            const float* __restrict__ pl2,
                            const float* __restrict__ f1,
                            const float* __restrict__ f2,
                            int*   __restrict__ labels1,
                            float* __restrict__ pos_exp,
                            float* __restrict__ mask1,
                            float* __restrict__ neg_logits,
                            float* __restrict__ sums)
{
    int idx = blockIdx.x * blockDim.x + threadIdx.x;
    if (idx == 0) { sums[0] = 0.f; sums[1] = 0.f; sums[2] = 0.f; sums[3] = 0.f; }
    if (idx >= BATCH * PP) return;
    int s = idx / PP, p = idx % PP;

    const float* a = pl1 + (size_t)s * CC * PP + p;
    const float* b = pl2 + (size_t)s * CC * PP + p;

    // argmax over classes (strict > keeps first occurrence, matching jnp.argmax)
    float best = a[0]; int bi = 0;
    #pragma unroll
    for (int c = 1; c < CC; ++c) {
        float v = a[(size_t)c * PP];
        if (v > best) { best = v; bi = c; }
    }
    labels1[idx] = bi;

    // mask1 = argmax_c of boolean combo -> index of first class where true, else 0
    int mi = 0;
    #pragma unroll
    for (int c = 0; c < CC; ++c) {
        float x = a[(size_t)c * PP], y = b[(size_t)c * PP];
        if ((x < y) && (y > GAMMA_C)) { mi = c; break; }
    }
    mask1[idx] = (float)mi;

    // positive-pair logit
    const float* u = f1 + (size_t)s * DD * PP + p;
    const float* w = f2 + (size_t)s * DD * PP + p;
    float dot = 0.f;
    #pragma unroll 8
    for (int d = 0; d < DD; ++d) dot += u[(size_t)d * PP] * w[(size_t)d * PP];
    pos_exp[idx] = __expf(dot * INV_TEMP);

    neg_logits[idx] = 0.f;
}

// ------------------------------------------------------------------
// Kernel 2: negative pseudo-labels: argmax over classes of npl
// ------------------------------------------------------------------
__global__ void neglab_kernel(const float* __restrict__ npl,
                              int* __restrict__ neg_labels)
{
    int idx = blockIdx.x * blockDim.x + threadIdx.x;
    if (idx >= BATCH * NN) return;
    int s = idx / NN, n = idx % NN;
    const float* a = npl + (size_t)s * CC * NN + n;
    float best = a[0]; int bi = 0;
    #pragma unroll
    for (int c = 1; c < CC; ++c) {
        float v = a[(size_t)c * NN];
        if (v > best) { best = v; bi = c; }
    }
    neg_labels[idx] = bi;
}

// ------------------------------------------------------------------
// Kernel 3: main WMMA GEMM + masked exp row-sum.
// Block tile: 48 (P) x 384 (N); 8 waves. Each wave holds a 3x3 grid of
// 16x16 WMMA C-tiles: 3 A fragments (3 P-tiles, from LDS, pre-paired
// float2) x 3 B fragments (from global) -> 9 WMMAs per K-step, ~1
// memory op per WMMA and 3x less L2 B-traffic than 1 P-tile/block.
// f32 exact path: v_wmma_f32_16x16x4_f32, K-loop of 32 steps.
// ------------------------------------------------------------------
__global__ __launch_bounds__(256)
void score_kernel(const float* __restrict__ f1,     // [B, D, PP]
                  const float* __restrict__ nf,     // [B, D, NN]
                  const int*   __restrict__ labels1,
                  const int*   __restrict__ neg_labels,
                  float* __restrict__ neg_logits)
{
    // A tile pre-paired along K, 48 columns interleaved:
    // As2[kp*48 + r] = {A[2kp][p0+r], A[2kp+1][p0+r]},  r in [0,48)
    __shared__ v2f   As2[(DD / 2) * PBLK];   // 24 KB
    __shared__ float rowsum[PBLK];

    const int s   = blockIdx.z;
    const int p0  = blockIdx.y * PBLK;
    const int n0  = blockIdx.x * NBLK;
    const int tid  = threadIdx.x;
    const int lane = tid & 31;
    const int wave = tid >> 5;

    const float* Ab = f1 + (size_t)s * DD * PP + p0;
    const float* Bb = nf + (size_t)s * DD * NN + n0 + wave * 48;

    for (int i = tid; i < (DD / 2) * PBLK; i += 256) {
        int kp = i / PBLK, r = i % PBLK;
        v2f t;
        t.x = Ab[(size_t)(2 * kp)     * PP + r];
        t.y = Ab[(size_t)(2 * kp + 1) * PP + r];
        As2[i] = t;                          // ds_store_b64
    }
    for (int i = tid; i < PBLK; i += 256) rowsum[i] = 0.f;
    __syncthreads();

    // ISA 32-bit A/B fragment layout: m/n = lane&15, k-pair base = 2*(lane>>4)
    const int mn   = lane & 15;
    const int half = lane >> 4;

    v8f acc[3][3];
    #pragma unroll
    for (int t = 0; t < 3; ++t)
        #pragma unroll
        for (int u = 0; u < 3; ++u)
            acc[t][u] = (v8f){};

    #pragma unroll 2
    for (int k = 0; k < DD; k += 4) {
        const int pairBase = ((k >> 1) + half) * PBLK + mn;
        v2f a0 = As2[pairBase];              // P-tile 0   (ds_load_b64)
        v2f a1 = As2[pairBase + 16];         // P-tile 1
        v2f a2 = As2[pairBase + 32];         // P-tile 2
        const float* Brow = Bb + (size_t)(k + 2 * half) * NN + mn;
        v2f b0, b1, b2;
        b0.x = Brow[0];        b0.y = Brow[NN];
        b1.x = Brow[16];       b1.y = Brow[NN + 16];
        b2.x = Brow[32];       b2.y = Brow[NN + 32];

        acc[0][0] = __builtin_amdgcn_wmma_f32_16x16x4_f32(false, a0, false, b0, (short)0, acc[0][0], false, false);
        acc[0][1] = __builtin_amdgcn_wmma_f32_16x16x4_f32(false, a0, false, b1, (short)0, acc[0][1], false, false);
        acc[0][2] = __builtin_amdgcn_wmma_f32_16x16x4_f32(false, a0, false, b2, (short)0, acc[0][2], false, false);
        acc[1][0] = __builtin_amdgcn_wmma_f32_16x16x4_f32(false, a1, false, b0, (short)0, acc[1][0], false, false);
        acc[1][1] = __builtin_amdgcn_wmma_f32_16x16x4_f32(false, a1, false, b1, (short)0, acc[1][1], false, false);
        acc[1][2] = __builtin_amdgcn_wmma_f32_16x16x4_f32(false, a1, false, b2, (short)0, acc[1][2], false, false);
        acc[2][0] = __builtin_amdgcn_wmma_f32_16x16x4_f32(false, a2, false, b0, (short)0, acc[2][0], false, false);
        acc[2][1] = __builtin_amdgcn_wmma_f32_16x16x4_f32(false, a2, false, b1, (short)0, acc[2][1], false, false);
        acc[2][2] = __builtin_amdgcn_wmma_f32_16x16x4_f32(false, a2, false, b2, (short)0, acc[2][2], false, false);
    }

    // C/D layout: element j of this lane is (row m = j + 8*half, col n = mn).
    // The 3 B-subtiles of row-group t hit the same 16 rows -> sum in-register
    // before the cross-lane butterfly.
    const size_t nrow = (size_t)s * NN + n0 + wave * 48 + mn;
    const int nl0 = neg_labels[nrow];
    const int nl1 = neg_labels[nrow + 16];
    const int nl2 = neg_labels[nrow + 32];

    #pragma unroll
    for (int t = 0; t < 3; ++t) {
        const int* labt = labels1 + (size_t)s * PP + p0 + t * 16 + half * 8;
        #pragma unroll
        for (int j = 0; j < 8; ++j) {
            int lj = labt[j];
            float v = 0.f;
            v += (lj != nl0) ? __expf(acc[t][0][j] * INV_TEMP) : 0.f;
            v += (lj != nl1) ? __expf(acc[t][1][j] * INV_TEMP) : 0.f;
            v += (lj != nl2) ? __expf(acc[t][2][j] * INV_TEMP) : 0.f;
            // butterfly over the 16 columns of this half-wave (xor <= 8 stays in half)
            SWZ_ADD(v, 0x041F);   // xor 1
            SWZ_ADD(v, 0x081F);   // xor 2
            SWZ_ADD(v, 0x101F);   // xor 4
            SWZ_ADD(v, 0x201F);   // xor 8
            if (mn == 0)
                atomicAdd(&rowsum[t * 16 + half * 8 + j], v);   // ds_add_f32
        }
    }
    __syncthreads();

    if (tid < PBLK)
        atomicAdd(&neg_logits[(size_t)s * PP + p0 + tid], rowsum[tid]);
}

// ------------------------------------------------------------------
// Kernel 4: per-anchor loss, accumulate weighted sum + mask sum
// ------------------------------------------------------------------
__global__ void loss_kernel(const float* __restrict__ neg_logits,
                            const float* __restrict__ pos_exp,
                            const float* __restrict__ mask1,
                            float* __restrict__ sums)
{
    int idx = blockIdx.x * blockDim.x + threadIdx.x;
    if (idx >= BATCH * PP) return;
    int s = idx / PP;
    float pe = pos_exp[idx];
    float nl = neg_logits[idx];
    float lossn = -logf(pe / (pe + nl + EPS_C) + EPS_C);
    float m = mask1[idx];
    atomicAdd(&sums[s * 2 + 0], m * lossn);
    atomicAdd(&sums[s * 2 + 1], m);
}

// ------------------------------------------------------------------
// Kernel 5: scalar output
// ------------------------------------------------------------------
__global__ void out_kernel(const float* __restrict__ sums, float* __restrict__ out)
{
    if (blockIdx.x == 0 && threadIdx.x == 0) {
        float l0 = sums[0] / (sums[1] + EPS_C);
        float l1 = sums[2] / (sums[3] + EPS_C);
        out[0] = 0.1f * (l0 + l1);
    }
}

// ------------------------------------------------------------------
extern "C" void kernel_launch(void* const* d_in, const int* in_sizes, int n_in,
                              void* d_out, int out_size, void* d_ws, size_t ws_size,
                              hipStream_t stream)
{
    (void)in_sizes; (void)n_in; (void)out_size; (void)ws_size;

    const float* feats_view1 = (const float*)d_in[0];  // [2,128,240,240]
    const float* pos_feats1  = (const float*)d_in[1];  // [2,128,60,60]
    const float* pos_feats2  = (const float*)d_in[2];  // [2,128,60,60]
    const float* plog1       = (const float*)d_in[3];  // [2,19,60,60]
    const float* plog2       = (const float*)d_in[4];  // [2,19,60,60]
    const float* neg_plog    = (const float*)d_in[5];  // [2,19,240,240]
    float* out = (float*)d_out;

    // workspace carve-up
    char* ws = (char*)d_ws;
    int*   neg_labels = (int*)ws;                                  // 2*57600
    int*   labels1    = (int*)(ws + (size_t)BATCH * NN * 4);       // 2*3600
    float* pos_exp    = (float*)((char*)labels1 + (size_t)BATCH * PP * 4);
    float* mask1v     = (float*)((char*)pos_exp + (size_t)BATCH * PP * 4);
    float* neg_logits = (float*)((char*)mask1v  + (size_t)BATCH * PP * 4);
    float* sums       = (float*)((char*)neg_logits + (size_t)BATCH * PP * 4); // 4 floats

    const int tpb = 256;

    prep_kernel<<<(BATCH * PP + tpb - 1) / tpb, tpb, 0, stream>>>(
        plog1, plog2, pos_feats1, pos_feats2,
        labels1, pos_exp, mask1v, neg_logits, sums);

    neglab_kernel<<<(BATCH * NN + tpb - 1) / tpb, tpb, 0, stream>>>(
        neg_plog, neg_labels);

    dim3 grid(NN / NBLK, PP / PBLK, BATCH);   // 150 x 75 x 2
    score_kernel<<<grid, tpb, 0, stream>>>(
        pos_feats1, feats_view1, labels1, neg_labels, neg_logits);

    loss_kernel<<<(BATCH * PP + tpb - 1) / tpb, tpb, 0, stream>>>(
        neg_logits, pos_exp, mask1v, sums);

    out_kernel<<<1, 32, 0, stream>>>(sums, out);
}